// NPNASGcnPredictorAgent_34256659153349
// MI455X (gfx1250) — compile-verified
//
#include <hip/hip_runtime.h>

// NPNAS GCN predictor for MI455X (gfx1250), wave32.
//
// Per layer l:
//   1) k_gemm_dual<K> : hF = h @ Wf_l, hB = h @ Wb_l  (v_wmma_f32_16x16x4_f32)
//   2) k_init         : acc = bias + h*dinv^2         (self-loop term)
//   3) k_scatter      : acc[d] += hF[g]*dinv[g]*dinv[d] over edges (f32 atomics)
//   4) k_combine      : h = 0.5*(relu(acc1)+relu(acc2)) in place
// Head: fc1/fc2 collapsed (no nonlinearity between them): weff = W1@W2,
// c0 = b2 + b1.W2, then fused mean-pool + dot in k_head.
//
// App roofline: edge gather/scatter moves ~14 GB (6 passes x 2M edges x 144
// f32 gathered + same volume of f32 atomics) -> ~0.6 ms at 23.3 TB/s; the
// GEMMs (8.4e10 f32 FLOPs) ride WMMA and stream 288 MB in / 576 MB out per
// layer. Full f32 keeps reference precision; compute is not the bottleneck.

#define HDIM 144
#define LHDIM 128

typedef float v2f __attribute__((ext_vector_type(2)));
typedef float v8f __attribute__((ext_vector_type(8)));

// ---------------- degree / normalization ----------------

__global__ void k_zero(float* __restrict__ p, int n) {
  int i = blockIdx.x * blockDim.x + threadIdx.x;
  if (i < n) p[i] = 0.0f;
}

__global__ void k_count(const int* __restrict__ src, const int* __restrict__ dst,
                        float* __restrict__ din, float* __restrict__ dout, int E) {
  int e = blockIdx.x * blockDim.x + threadIdx.x;
  if (e < E) {
    atomicAdd(&din[dst[e]], 1.0f);   // in-degree
    atomicAdd(&dout[src[e]], 1.0f);  // out-degree
  }
}

__global__ void k_rsqrt(float* __restrict__ p, int n) {
  int i = blockIdx.x * blockDim.x + threadIdx.x;
  if (i < n) p[i] = rsqrtf(p[i] + 1.0f);  // +1 self-loop; always >= 1
}

// ---------------- dual GEMM via f32 WMMA ----------------
// Block = 288 threads = 9 waves; wave w owns columns [16w,16w+16) of H=144.
// Block owns MT 16-row subtiles; A staged in LDS once, each B fragment load
// feeds 2*MT WMMAs. Full tiles take a uniform fast path: unconditional
// vectorized LDS fill and unconditional stores with immediate offsets.

template <int K, int KP, int MT>
__global__ __launch_bounds__(288) void k_gemm_dual(
    const float* __restrict__ X,
    const float* __restrict__ Wf, const float* __restrict__ Wb,
    float* __restrict__ outF, float* __restrict__ outB, int nrows) {
  constexpr int STR = KP + 4;     // row stride: 16B-aligned rows, conflict-free
  constexpr int MROWS = 16 * MT;
  __shared__ float As[MROWS * STR];
  const int row0 = blockIdx.x * MROWS;
  const int tid = threadIdx.x;
  const bool full = (row0 + MROWS) <= nrows;   // wave-uniform

  if constexpr (K == KP) {
    if (full) {
      // contiguous rows: float4 fill (global_load_b128 + ds_store_b128)
      constexpr int KV = KP / 4;
      for (int i = tid; i < MROWS * KV; i += 288) {
        int m = i / KV, k4 = i - m * KV;
        float4 v = ((const float4*)(X + (size_t)(row0 + m) * K))[k4];
        *(float4*)(&As[m * STR + 4 * k4]) = v;
      }
    } else {
      for (int i = tid; i < MROWS * KP; i += 288) {
        int m = i / KP, k = i - m * KP;
        int r = row0 + m;
        As[m * STR + k] = (r < nrows) ? X[(size_t)r * K + k] : 0.0f;
      }
    }
  } else {
    // K not multiple of 4 (layer 0): clamp+select, no OOB reads
    for (int i = tid; i < MROWS * KP; i += 288) {
      int m = i / KP, k = i - m * KP;
      int r = row0 + m;
      int kk = (k < K) ? k : 0;
      float v = 0.0f;
      if (r < nrows) v = X[(size_t)r * K + kk];
      As[m * STR + k] = (k < K) ? v : 0.0f;
    }
  }
  __syncthreads();

  const int wave = tid >> 5;
  const int lane = tid & 31;
  const int half = lane >> 4;   // 0: K pair {0,1}, 1: K pair {2,3}
  const int mm   = lane & 15;
  const int col  = wave * 16 + mm;

  const v8f vzero = {0.f, 0.f, 0.f, 0.f, 0.f, 0.f, 0.f, 0.f};
  v8f accF[MT];
  v8f accB[MT];
#pragma unroll
  for (int t = 0; t < MT; ++t) { accF[t] = vzero; accB[t] = vzero; }

  for (int k0 = 0; k0 < KP; k0 += 4) {
    const int ka = k0 + half * 2;
    v2f bf_, bb_;
    if constexpr (K == KP) {
      bf_.x = Wf[ka * HDIM + col];
      bf_.y = Wf[(ka + 1) * HDIM + col];
      bb_.x = Wb[ka * HDIM + col];
      bb_.y = Wb[(ka + 1) * HDIM + col];
    } else {
      int i0 = (ka < K) ? ka : 0;
      int i1 = (ka + 1 < K) ? (ka + 1) : 0;
      float f0 = Wf[i0 * HDIM + col];
      float f1 = Wf[i1 * HDIM + col];
      float g0 = Wb[i0 * HDIM + col];
      float g1 = Wb[i1 * HDIM + col];
      bf_.x = (ka < K) ? f0 : 0.0f;
      bf_.y = (ka + 1 < K) ? f1 : 0.0f;
      bb_.x = (ka < K) ? g0 : 0.0f;
      bb_.y = (ka + 1 < K) ? g1 : 0.0f;
    }
#pragma unroll
    for (int t = 0; t < MT; ++t) {
      v2f a;  // A 16x4: lane holds row mm of subtile t, K = ka, ka+1
      a.x = As[(t * 16 + mm) * STR + ka];
      a.y = As[(t * 16 + mm) * STR + ka + 1];
      accF[t] = __builtin_amdgcn_wmma_f32_16x16x4_f32(false, a, false, bf_,
                                                      (short)0, accF[t], false, false);
      accB[t] = __builtin_amdgcn_wmma_f32_16x16x4_f32(false, a, false, bb_,
                                                      (short)0, accB[t], false, false);
    }
  }

  // C/D layout: VGPR i -> row i (lanes 0-15) / row i+8 (lanes 16-31)
  if (full) {
#pragma unroll
    for (int t = 0; t < MT; ++t) {
      size_t base = (size_t)(row0 + t * 16 + half * 8) * HDIM + col;
      float* pF = outF + base;
      float* pB = outB + base;
#pragma unroll
      for (int i = 0; i < 8; ++i) {
        pF[i * HDIM] = accF[t][i];   // immediate offset i*576B
        pB[i * HDIM] = accB[t][i];
      }
    }
  } else {
#pragma unroll
    for (int t = 0; t < MT; ++t) {
#pragma unroll
      for (int i = 0; i < 8; ++i) {
        int r = row0 + t * 16 + i + half * 8;
        if (r < nrows) {
          outF[(size_t)r * HDIM + col] = accF[t][i];
          outB[(size_t)r * HDIM + col] = accB[t][i];
        }
      }
    }
  }
}

// ---------------- aggregation ----------------

__global__ void k_init(const float* __restrict__ h, const float* __restrict__ dinv,
                       const float* __restrict__ bias, float* __restrict__ acc,
                       int total) {
  int i = blockIdx.x * blockDim.x + threadIdx.x;
  if (i >= total) return;
  int node = i / HDIM;
  int c = i - node * HDIM;
  float dv = dinv[node];
  acc[i] = bias[c] + h[i] * dv * dv;  // bias + self-loop contribution
}

__global__ __launch_bounds__(256) void k_scatter(
    const int* __restrict__ gi, const int* __restrict__ di,
    const float* __restrict__ dinv, const float* __restrict__ hin,
    float* __restrict__ acc, int E) {
  int e = blockIdx.x * 8 + (threadIdx.x >> 5);  // one wave per edge
  if (e >= E) return;
  int lane = threadIdx.x & 31;
  int g = gi[e];
  int d = di[e];
  float w = dinv[g] * dinv[d];
  const float* hp = hin + (size_t)g * HDIM;
  float* ap = acc + (size_t)d * HDIM;
  for (int c = lane; c < HDIM; c += 32)
    atomicAdd(&ap[c], hp[c] * w);
}

__global__ void k_combine(float* __restrict__ a1, const float* __restrict__ a2,
                          int total) {
  int i = blockIdx.x * blockDim.x + threadIdx.x;
  if (i >= total) return;
  a1[i] = 0.5f * (fmaxf(a1[i], 0.0f) + fmaxf(a2[i], 0.0f));
}

// ---------------- head (fc1/fc2 collapsed) ----------------

__global__ void k_weff(const float* __restrict__ W1, const float* __restrict__ b1,
                       const float* __restrict__ W2, const float* __restrict__ b2,
                       float* __restrict__ weff) {
  int c = threadIdx.x;
  if (c < HDIM) {
    float s = 0.0f;
    for (int j = 0; j < LHDIM; ++j) s += W1[c * LHDIM + j] * W2[j];
    weff[c] = s;
  } else if (c == HDIM) {
    float s = b2[0];
    for (int j = 0; j < LHDIM; ++j) s += b1[j] * W2[j];
    weff[HDIM] = s;  // constant term
  }
}

__global__ __launch_bounds__(256) void k_head(const float* __restrict__ h,
                                              const float* __restrict__ weff,
                                              float* __restrict__ out,
                                              int G, int npg) {
  int g = blockIdx.x * 8 + (threadIdx.x >> 5);  // one wave per graph
  if (g >= G) return;
  int lane = threadIdx.x & 31;
  int tot = npg * HDIM;
  size_t base = (size_t)g * npg * HDIM;
  float s = 0.0f;
  for (int t = lane; t < tot; t += 32) {
    int c = t % HDIM;
    s += h[base + t] * weff[c];
  }
  for (int o = 16; o > 0; o >>= 1) s += __shfl_down(s, o, 32);
  if (lane == 0) out[g] = weff[HDIM] + s / (float)npg;  // mean-pool then head
}

// ---------------- launcher ----------------

extern "C" void kernel_launch(void* const* d_in, const int* in_sizes, int n_in,
                              void* d_out, int out_size, void* d_ws, size_t ws_size,
                              hipStream_t stream) {
  const int DIN = 5;
  const float* x  = (const float*)d_in[0];
  const int*   ei = (const int*)d_in[1];
  const int N = in_sizes[0] / DIN;
  const int E = in_sizes[1] / 2;
  const int G = out_size;
  const int npg = N / G;  // 10 nodes per graph, batch is sorted i -> i/npg
  const int* src = ei;
  const int* dst = ei + E;

  const float* Wf[3] = {(const float*)d_in[3], (const float*)d_in[7],  (const float*)d_in[11]};
  const float* bf[3] = {(const float*)d_in[4], (const float*)d_in[8],  (const float*)d_in[12]};
  const float* Wb[3] = {(const float*)d_in[5], (const float*)d_in[9],  (const float*)d_in[13]};
  const float* bb[3] = {(const float*)d_in[6], (const float*)d_in[10], (const float*)d_in[14]};
  const float* W1 = (const float*)d_in[15];
  const float* b1 = (const float*)d_in[16];
  const float* W2 = (const float*)d_in[17];
  const float* b2 = (const float*)d_in[18];
  float* out = (float*)d_out;

  // workspace layout (floats): weff[256] | dinv_in[N] | dinv_out[N] | B0..B3[N*H]
  float* wsf      = (float*)d_ws;
  float* weff     = wsf;
  float* dinv_in  = wsf + 256;
  float* dinv_out = dinv_in + N;
  size_t NH = (size_t)N * HDIM;
  float* B0 = dinv_out + N;
  float* B1 = B0 + NH;
  float* B2 = B1 + NH;
  float* B3 = B2 + NH;

  const int totNH = N * HDIM;
  const int TB = 256;

  // degrees -> dinv (shared by all layers)
  k_zero <<<(2 * N + TB - 1) / TB, TB, 0, stream>>>(dinv_in, 2 * N);
  k_count<<<(E + TB - 1) / TB, TB, 0, stream>>>(src, dst, dinv_in, dinv_out, E);
  k_rsqrt<<<(2 * N + TB - 1) / TB, TB, 0, stream>>>(dinv_in, 2 * N);
  k_weff <<<1, 160, 0, stream>>>(W1, b1, W2, b2, weff);

  constexpr int MT = 4;
  constexpr int MROWS = 16 * MT;
  const int gemmGrid = (N + MROWS - 1) / MROWS;
  const int nhGrid   = (totNH + TB - 1) / TB;
  const int edgeGrid = (E + 7) / 8;

  for (int l = 0; l < 3; ++l) {
    if (l == 0) {
      k_gemm_dual<5, 8, MT><<<gemmGrid, 288, 0, stream>>>(x, Wf[0], Wb[0], B1, B2, N);
    } else {
      k_gemm_dual<144, 144, MT><<<gemmGrid, 288, 0, stream>>>(B0, Wf[l], Wb[l], B1, B2, N);
    }
    // forward direction (A): acc1 in B0
    k_init   <<<nhGrid, TB, 0, stream>>>(B1, dinv_in, bf[l], B0, totNH);
    k_scatter<<<edgeGrid, TB, 0, stream>>>(src, dst, dinv_in, B1, B0, E);
    // backward direction (A^T): acc2 in B3
    k_init   <<<nhGrid, TB, 0, stream>>>(B2, dinv_out, bb[l], B3, totNH);
    k_scatter<<<edgeGrid, TB, 0, stream>>>(dst, src, dinv_out, B2, B3, E);
    // h <- 0.5*(relu+relu), in place into B0
    k_combine<<<nhGrid, TB, 0, stream>>>(B0, B3, totNH);
  }

  k_head<<<(G + 7) / 8, TB, 0, stream>>>(B0, weff, out, G, npg);
}